// ComplexNet_2207613190838
// MI455X (gfx1250) — compile-verified
//
#include <hip/hip_runtime.h>
#include <cstdint>
#include <cstddef>

typedef __attribute__((ext_vector_type(2))) float v2f;
typedef __attribute__((ext_vector_type(8))) float v8f;

#define TPB 256

// ---------------- degree / norm ----------------
__global__ void deg_init_kernel(float* __restrict__ deg, int n) {
  int i = blockIdx.x * blockDim.x + threadIdx.x;
  if (i < n) deg[i] = 1.0f;  // self-loop contribution
}

__global__ void deg_accum_kernel(const long long* __restrict__ ei, long long e_cnt,
                                 float* __restrict__ deg) {
  long long e = (long long)blockIdx.x * blockDim.x + threadIdx.x;
  if (e < e_cnt) atomicAdd(&deg[ei[e_cnt + e]], 1.0f);  // dst-counted degree
}

__global__ void deg_rsqrt_kernel(float* __restrict__ deg, int n) {
  int i = blockIdx.x * blockDim.x + threadIdx.x;
  if (i < n) deg[i] = rsqrtf(deg[i]);
}

__global__ void zero_kernel(float* __restrict__ p, long long n) {
  long long i = (long long)blockIdx.x * blockDim.x + threadIdx.x;
  if (i < n) p[i] = 0.0f;
}

// ---------------- WMMA GEMM: H[N x NC] = X[N x K] @ W[K x nsrc] (zero-pad to NC) ----
// One wave computes a 16x16 output tile via V_WMMA_F32_16X16X4_F32.
// B is staged in LDS once per block (W1 = 64KB <= 320KB WGP LDS).
template<int K, int NC>
__global__ void gemm_wmma_kernel(const float* __restrict__ X, const float* __restrict__ W,
                                 int nsrc, float* __restrict__ H, int nrows, int rowTiles) {
  __shared__ float Blds[K * NC];
  for (int i = threadIdx.x; i < K * NC; i += blockDim.x) {
    int k = i / NC, nn = i % NC;
    Blds[i] = (nn < nsrc) ? W[k * nsrc + nn] : 0.0f;
  }
  __syncthreads();

  constexpr int CT = NC / 16;
  int wv = blockIdx.x * (blockDim.x >> 5) + (threadIdx.x >> 5);
  if (wv >= rowTiles * CT) return;            // wave-uniform guard: EXEC stays all-1s
  int rowBase = (wv / CT) * 16;
  int colBase = (wv % CT) * 16;
  int lane = threadIdx.x & 31;
  int half = lane >> 4;                       // lanes 16-31 carry K+2,K+3 halves
  int l16  = lane & 15;

  int arow = rowBase + l16;
  if (arow >= nrows) arow = nrows - 1;        // clamp (N % 16 == 0 -> never taken)
  const float* xr = X + (size_t)arow * K;

  v8f c = {};
  for (int k = 0; k < K; k += 4) {
    int ka = k + half * 2;
    v2f a, b;
    a.x = xr[ka + 0];                          // A 16x4: lane half selects K pair
    a.y = xr[ka + 1];
    b.x = Blds[(ka + 0) * NC + colBase + l16]; // B 4x16: N striped over lanes
    b.y = Blds[(ka + 1) * NC + colBase + l16];
    c = __builtin_amdgcn_wmma_f32_16x16x4_f32(false, a, false, b, (short)0, c,
                                              false, false);
  }

  // D: VGPR r -> row rowBase + r (+8 for lanes 16-31)
  float* hb = H + (size_t)(rowBase + half * 8) * NC + colBase + l16;
  if (rowBase + 16 <= nrows) {                // wave-uniform: full tile in-bounds
#pragma unroll
    for (int r = 0; r < 8; ++r)
      hb[(size_t)r * NC] = c[r];              // unguarded -> clause-able stores
  } else {
#pragma unroll
    for (int r = 0; r < 8; ++r) {
      int row = rowBase + half * 8 + r;
      if (row < nrows)
        H[(size_t)row * NC + colBase + l16] = c[r];
    }
  }
}

// ---------------- edge scatter: agg[dst] += H[src] * dis[src]*dis[dst] -------------
template<int F, int S>
__global__ void scatter_kernel(const long long* __restrict__ ei, long long e_cnt,
                               const float* __restrict__ H, const float* __restrict__ dis,
                               float* __restrict__ agg) {
  long long t = (long long)blockIdx.x * blockDim.x + threadIdx.x;
  long long e = t / S;
  int f = (int)(t - e * S);
  if (e >= e_cnt) return;
  if (F < S && f >= F) return;
  long long s = ei[e];
  long long d = ei[e_cnt + e];
  float nrm = dis[s] * dis[d];
  atomicAdd(&agg[(size_t)d * S + f], H[(size_t)s * S + f] * nrm);
}

// ---------------- self-loop + bias (+ ReLU), in place on agg -----------------------
template<int F, int S, bool RELU>
__global__ void finalize_kernel(float* __restrict__ agg, const float* __restrict__ H,
                                const float* __restrict__ dis,
                                const float* __restrict__ bias, int n) {
  long long i = (long long)blockIdx.x * blockDim.x + threadIdx.x;
  long long node = i / S;
  int f = (int)(i - node * S);
  if (node >= n) return;
  if (F < S && f >= F) { agg[i] = 0.0f; return; }   // keep pad columns zeroed
  float di = dis[node];
  float v = agg[i] + H[i] * (di * di) + bias[f];
  if (RELU) v = fmaxf(v, 0.0f);
  agg[i] = v;
}

// ---------------- log_softmax over 11 classes (stride-16 input) --------------------
__global__ void logsoftmax_kernel(const float* __restrict__ A, float* __restrict__ out,
                                  int n) {
  int i = blockIdx.x * blockDim.x + threadIdx.x;
  if (i >= n) return;
  const float* v = A + (size_t)i * 16;
  float m = v[0];
#pragma unroll
  for (int c = 1; c < 11; ++c) m = fmaxf(m, v[c]);
  float s = 0.0f;
#pragma unroll
  for (int c = 0; c < 11; ++c) s += expf(v[c] - m);
  float ls = logf(s);
#pragma unroll
  for (int c = 0; c < 11; ++c) out[(size_t)i * 11 + c] = v[c] - m - ls;
}

extern "C" void kernel_launch(void* const* d_in, const int* in_sizes, int n_in,
                              void* d_out, int out_size, void* d_ws, size_t ws_size,
                              hipStream_t stream) {
  (void)n_in; (void)out_size; (void)ws_size;
  const float*     x  = (const float*)d_in[0];
  const long long* ei = (const long long*)d_in[1];   // int64 edge_index [2, E]
  const float* W1 = (const float*)d_in[2];
  const float* b1 = (const float*)d_in[3];
  const float* W2 = (const float*)d_in[4];
  const float* b2 = (const float*)d_in[5];
  const float* W3 = (const float*)d_in[6];
  const float* b3 = (const float*)d_in[7];
  float* out = (float*)d_out;

  const int N = in_sizes[0] / 512;                   // 100000
  const long long E = (long long)in_sizes[1] / 2;    // 3200000

  // workspace: 65*N floats (26 MB), overlapped across layers
  float* ws   = (float*)d_ws;
  float* dis  = ws;                  // [N]      deg -> deg^{-1/2}
  float* buf1 = ws + N;              // [32N]    h1 raw; later h2 | agg2
  float* buf2 = buf1 + 32LL * N;     // [32N]    agg1/act1; later h3 | agg3
  float* h2   = buf1;
  float* agg2 = buf1 + 16LL * N;
  float* h3   = buf2;
  float* agg3 = buf2 + 16LL * N;

  const int rowTiles = (N + 15) / 16;

  // degrees + symmetric norm
  deg_init_kernel<<<(N + TPB - 1) / TPB, TPB, 0, stream>>>(dis, N);
  deg_accum_kernel<<<(int)((E + TPB - 1) / TPB), TPB, 0, stream>>>(ei, E, dis);
  deg_rsqrt_kernel<<<(N + TPB - 1) / TPB, TPB, 0, stream>>>(dis, N);

  // ---- layer 1: 512 -> 32, ReLU ----
  {
    int waves = rowTiles * 2;  // 32 cols = 2 col tiles
    gemm_wmma_kernel<512, 32><<<(waves + 7) / 8, TPB, 0, stream>>>(x, W1, 32, buf1, N, rowTiles);
    zero_kernel<<<(int)((32LL * N + TPB - 1) / TPB), TPB, 0, stream>>>(buf2, 32LL * N);
    long long tot = E * 32;
    scatter_kernel<32, 32><<<(int)((tot + TPB - 1) / TPB), TPB, 0, stream>>>(ei, E, buf1, dis, buf2);
    finalize_kernel<32, 32, true><<<(int)((32LL * N + TPB - 1) / TPB), TPB, 0, stream>>>(buf2, buf1, dis, b1, N);
  }
  // ---- layer 2: 32 -> 16, ReLU ----
  {
    gemm_wmma_kernel<32, 16><<<(rowTiles + 7) / 8, TPB, 0, stream>>>(buf2, W2, 16, h2, N, rowTiles);
    zero_kernel<<<(int)((16LL * N + TPB - 1) / TPB), TPB, 0, stream>>>(agg2, 16LL * N);
    long long tot = E * 16;
    scatter_kernel<16, 16><<<(int)((tot + TPB - 1) / TPB), TPB, 0, stream>>>(ei, E, h2, dis, agg2);
    finalize_kernel<16, 16, true><<<(int)((16LL * N + TPB - 1) / TPB), TPB, 0, stream>>>(agg2, h2, dis, b2, N);
  }
  // ---- layer 3: 16 -> 11 (padded to 16), no ReLU ----
  {
    gemm_wmma_kernel<16, 16><<<(rowTiles + 7) / 8, TPB, 0, stream>>>(agg2, W3, 11, h3, N, rowTiles);
    zero_kernel<<<(int)((16LL * N + TPB - 1) / TPB), TPB, 0, stream>>>(agg3, 16LL * N);
    long long tot = E * 16;
    scatter_kernel<11, 16><<<(int)((tot + TPB - 1) / TPB), TPB, 0, stream>>>(ei, E, h3, dis, agg3);
    finalize_kernel<11, 16, false><<<(int)((16LL * N + TPB - 1) / TPB), TPB, 0, stream>>>(agg3, h3, dis, b3, N);
  }

  logsoftmax_kernel<<<(N + TPB - 1) / TPB, TPB, 0, stream>>>(agg3, out, N);
}